// feature_catch_77137612636722
// MI455X (gfx1250) — compile-verified
//
#include <hip/hip_runtime.h>
#include <stdint.h>

// Problem constants (from reference)
#define BATCH      65536
#define L_SIG      9
#define W_LEN      30
#define D_WIN      10
#define NWIN       3
#define NPAIR      36
#define IN_PER_B   (L_SIG * W_LEN)        // 270
#define STAGE      288                    // 9 * 32, padded staging region
#define OUT_PER_B  (6 * NPAIR * NWIN)     // 648 (= 162 float4)
#define WAVES_PB   8
#define THREADS    (WAVES_PB * 32)

typedef float v2f __attribute__((ext_vector_type(2)));
typedef float v8f __attribute__((ext_vector_type(8)));

__device__ __forceinline__ uint32_t lds_off(const void* p) {
    // addrspacecast(LDS->flat): low 32 bits hold the wave-relative LDS byte
    // offset; hardware uses addr[31:0] for the LDS aperture.
    return (uint32_t)(uint64_t)(uintptr_t)p;
}

__device__ __forceinline__ void async_load_b32(uint32_t loff, const float* g) {
    unsigned long long ga = (unsigned long long)(const void*)g;
    asm volatile("global_load_async_to_lds_b32 %0, %1, off"
                 :: "v"(loff), "v"(ga) : "memory");
}

__device__ __forceinline__ void async_store_b128(float* g, uint32_t loff) {
    unsigned long long ga = (unsigned long long)(void*)g;
    asm volatile("global_store_async_from_lds_b128 %0, %1, off"
                 :: "v"(ga), "v"(loff) : "memory");
}

__device__ __forceinline__ void wait_async0() {
    asm volatile("s_wait_asynccnt 0x0" ::: "memory");
}

__device__ __forceinline__ void lds_fence_wave() {
    // Wave-internal LDS ordering: LDS ops from one wave execute in order;
    // after DScnt==0 all prior DS stores are committed and visible to all
    // lanes of this wave. (All inter-lane traffic here is per-wave.)
    asm volatile("s_wait_dscnt 0x0" ::: "memory");
}

__device__ __forceinline__ void pair_ij(int p, int& i, int& j) {
    // triu_indices(9, k=1) row-major: rows of length 8,7,...,1
    int ii = 0, rem = p, row = 8;
    while (rem >= row) { rem -= row; row--; ii++; }
    i = ii; j = ii + 1 + rem;
}

// Exact closed form of betainc(4, 0.5, clip(1 - r^2, 0, 1)):
//   p = 1 - (35/16) * (t - t^3 + (3/5) t^5 - t^7/7),  t = |r| clamped to 1
__device__ __forceinline__ float pval_from_G(const float* gb, int i, int j) {
    float gii = gb[i * 16 + i];
    float gjj = gb[j * 16 + j];
    float gij = gb[j * 16 + i];                    // G[i][j], i < j <= 8
    float r   = gij * __builtin_amdgcn_rsqf(gii * gjj);
    float t   = fminf(fabsf(r), 1.0f);
    float t2  = t * t;
    float poly = t * (1.0f + t2 * (-1.0f + t2 * (0.6f - t2 * (1.0f / 7.0f))));
    return 1.0f - 2.1875f * poly;                  // 35/16 = 2.1875
}

__global__ __launch_bounds__(THREADS)
void feature_fused_kernel(const float* __restrict__ x, float* __restrict__ out) {
    __shared__ alignas(16) float sbuf[WAVES_PB][STAGE];     // staged input rows
    __shared__ alignas(16) float gbuf[WAVES_PB][16 * 16];   // Gram scratch
    __shared__ alignas(16) float obuf[WAVES_PB][OUT_PER_B]; // output accumulation

    const int lane  = threadIdx.x & 31;
    const int wv    = threadIdx.x >> 5;
    const int batch = blockIdx.x * WAVES_PB + wv;

    const float* src = x + (size_t)batch * IN_PER_B;

    // ---- Stage this wave's batch row into LDS: 9 back-to-back async copies ----
    #pragma unroll
    for (int it = 0; it < STAGE / 32; ++it) {
        int t   = it * 32 + lane;
        int idx = (t < IN_PER_B) ? t : (IN_PER_B - 1);  // clamp: no OOB, uniform
        async_load_b32(lds_off(&sbuf[wv][t]), src + idx);
    }

    // ---- Zero the output buffer (covers zero padding of channels 2..5) ----
    {
        float4 z = make_float4(0.f, 0.f, 0.f, 0.f);
        #pragma unroll
        for (int it = 0; it < 5; ++it)
            *(float4*)&obuf[wv][(it * 32 + lane) * 4] = z;
        if (lane < 2)
            *(float4*)&obuf[wv][(160 + lane) * 4] = z;
    }

    // Pair assignments for this lane (fixed): p0 = lane, p1 = lane + 32 (if < 36)
    int i0, j0, i1 = 0, j1 = 1;
    pair_ij(lane, i0, j0);
    if (lane < 4) pair_ij(lane + 32, i1, j1);

    const int m     = lane & 15;          // Gram row/col this lane feeds
    const int khalf = lane >> 4;          // 0: K in {0,1}(+4t), 1: K in {2,3}(+4t)
    const int mm    = (m < L_SIG) ? m : 0;

    wait_async0();                        // staged data now resident in LDS

    for (int w = 0; w < NWIN; ++w) {
        // ---- Load this lane's (clamped) channel row for window w from LDS ----
        const float2* rp = (const float2*)&sbuf[wv][mm * 30 + w * 10]; // even off
        float2 q0 = rp[0], q1 = rp[1], q2 = rp[2], q3 = rp[3], q4 = rp[4];
        float e0 = q0.x, e1 = q0.y, e2 = q1.x, e3 = q1.y, e4 = q2.x,
              e5 = q2.y, e6 = q3.x, e7 = q3.y, e8 = q4.x, e9 = q4.y;

        float sum  = ((e0 + e1) + (e2 + e3)) + ((e4 + e5) + (e6 + e7)) + (e8 + e9);
        float mean = sum * 0.1f;
        float dl   = (10.f*e0 + 9.f*e1 + 8.f*e2 + 7.f*e3 + 6.f*e4 +
                       5.f*e5 + 4.f*e6 + 3.f*e7 + 2.f*e8 + 1.f*e9) * (1.0f / 55.0f);

        float c0 = e0 - mean, c1 = e1 - mean, c2 = e2 - mean, c3 = e3 - mean,
              c4 = e4 - mean, c5 = e5 - mean, c6 = e6 - mean, c7 = e7 - mean,
              c8 = e8 - mean, c9 = e9 - mean;

        // ---- A (= B: Gram is symmetric, f32 A/B layouts mirror) chunks ----
        // lanes 0-15 hold K = 4t+{0,1}, lanes 16-31 hold K = 4t+{2,3}
        v2f a0, a1, a2;
        a0.x = khalf ? c2 : c0;  a0.y = khalf ? c3 : c1;
        a1.x = khalf ? c6 : c4;  a1.y = khalf ? c7 : c5;
        a2.x = khalf ? 0.f : c8; a2.y = khalf ? 0.f : c9;   // K=10,11 zero pad

        v8f acc = {};
        acc = __builtin_amdgcn_wmma_f32_16x16x4_f32(false, a0, false, a0,
                                                    (short)0, acc, false, false);
        acc = __builtin_amdgcn_wmma_f32_16x16x4_f32(false, a1, false, a1,
                                                    (short)0, acc, false, false);
        acc = __builtin_amdgcn_wmma_f32_16x16x4_f32(false, a2, false, a2,
                                                    (short)0, acc, false, false);

        // ---- Uniform G scatter: lane (m,khalf) owns rows 8*khalf..+7 of col m
        // C/D layout: VGPR r -> M = r + 8*khalf, N = m. Two b128 DS stores.
        float* gb = gbuf[wv];
        {
            float4 lo = make_float4(acc[0], acc[1], acc[2], acc[3]);
            float4 hi = make_float4(acc[4], acc[5], acc[6], acc[7]);
            *(float4*)&gb[m * 16 + 8 * khalf]     = lo;   // 16B aligned
            *(float4*)&gb[m * 16 + 8 * khalf + 4] = hi;
        }
        lds_fence_wave();   // scatter visible to all lanes of this wave

        float* ob = obuf[wv];

        // ---- Per-channel stats (lanes 0..8 own channel m = lane) ----
        if (lane < L_SIG) {
            float gii  = gb[lane * 16 + lane];             // sum(wc^2)
            float stdv = __builtin_amdgcn_sqrtf(gii * 0.1f);
            float zsv  = mean * __builtin_amdgcn_rsqf(gii * 0.1f);
            ob[2 * 108 + lane * 3 + w] = stdv;
            ob[3 * 108 + lane * 3 + w] = zsv;
            ob[4 * 108 + lane * 3 + w] = sum;
            ob[5 * 108 + lane * 3 + w] = dl;
        }

        // ---- 36 pair p-values (conv1 == conv2) ----
        {
            float pv = pval_from_G(gb, i0, j0);
            ob[0 * 108 + lane * 3 + w] = pv;
            ob[1 * 108 + lane * 3 + w] = pv;
        }
        if (lane < 4) {
            int p = lane + 32;
            float pv = pval_from_G(gb, i1, j1);
            ob[0 * 108 + p * 3 + w] = pv;
            ob[1 * 108 + p * 3 + w] = pv;
        }
        // No fence needed before next window: LDS ops from this wave execute
        // in order, so this window's gb reads precede next window's gb stores.
    }

    // ---- Flush 648 contiguous outputs with async LDS->global b128 stores ----
    lds_fence_wave();                       // obuf writes committed to LDS
    float* dstb = out + (size_t)batch * OUT_PER_B;
    #pragma unroll
    for (int it = 0; it < 5; ++it) {
        int t = it * 32 + lane;
        async_store_b128(dstb + t * 4, lds_off(&obuf[wv][t * 4]));
    }
    if (lane < 2) {
        int t = 160 + lane;
        async_store_b128(dstb + t * 4, lds_off(&obuf[wv][t * 4]));
    }
    wait_async0();   // (S_ENDPGM also implies wait-idle; explicit for clarity)
}

extern "C" void kernel_launch(void* const* d_in, const int* in_sizes, int n_in,
                              void* d_out, int out_size, void* d_ws, size_t ws_size,
                              hipStream_t stream) {
    const float* x = (const float*)d_in[0];
    float* out = (float*)d_out;
    dim3 grid(BATCH / WAVES_PB);   // 8192 blocks of 256 threads (8 waves)
    feature_fused_kernel<<<grid, THREADS, 0, stream>>>(x, out);
}